// DecoderStepModule_54606214201412
// MI455X (gfx1250) — compile-verified
//
#include <hip/hip_runtime.h>
#include <math.h>

typedef __attribute__((ext_vector_type(2))) float v2f;
typedef __attribute__((ext_vector_type(8))) float v8f;

#define Bn 16
#define Ln 4096
#define An 512
#define Hn 512
#define K2H 1024
#define NCHUNK 32
#define LCHUNK 128

// ---------------------------------------------------------------------------
// Kernel 1: dec[16][512] = cat(h,c)[16][1024] @ W[512][1024]^T + bias
// One wave (32 threads) per 16-wide N tile -> grid of 32 blocks.
// Uses V_WMMA_F32_16X16X4_F32 with 4 independent accumulator chains.
// ---------------------------------------------------------------------------
__global__ __launch_bounds__(32)
void dec_wmma_kernel(const float* __restrict__ h, const float* __restrict__ c,
                     const float* __restrict__ W, const float* __restrict__ bias,
                     float* __restrict__ dec) {
    const int lane = threadIdx.x;           // 0..31
    const int n0   = blockIdx.x * 16;       // N tile base in A-dim
    const int row  = lane & 15;             // M for A-operand, N-col for B-operand
    const int hi   = (lane >> 4) & 1;       // 0: K+0/K+1 ; 1: K+2/K+3
    const int koff = hi * 2;

    const int n    = n0 + row;              // column of dec this lane's B covers
    const float* __restrict__ Wn = W + (size_t)n * K2H;

    v8f acc0 = {}, acc1 = {}, acc2 = {}, acc3 = {};

    // ---- first half of K: X[:, 0:512] = h ----
    const float* __restrict__ Xrow = h + (size_t)row * Hn;
    #pragma unroll 4
    for (int kk = 0; kk < 512; kk += 16) {
        #pragma unroll
        for (int j = 0; j < 4; ++j) {
            const int k = kk + 4 * j;
            v2f a, b;
            a.x = Xrow[k + koff];
            a.y = Xrow[k + koff + 1];
            b.x = Wn[k + koff];
            b.y = Wn[k + koff + 1];
            v8f& acc = (j == 0) ? acc0 : (j == 1) ? acc1 : (j == 2) ? acc2 : acc3;
            acc = __builtin_amdgcn_wmma_f32_16x16x4_f32(
                false, a, false, b, (short)0, acc, false, false);
        }
    }
    // ---- second half of K: X[:, 512:1024] = c ----
    Xrow = c + (size_t)row * Hn;
    const float* __restrict__ Wn2 = Wn + 512;
    #pragma unroll 4
    for (int kk = 0; kk < 512; kk += 16) {
        #pragma unroll
        for (int j = 0; j < 4; ++j) {
            const int k = kk + 4 * j;
            v2f a, b;
            a.x = Xrow[k + koff];
            a.y = Xrow[k + koff + 1];
            b.x = Wn2[k + koff];
            b.y = Wn2[k + koff + 1];
            v8f& acc = (j == 0) ? acc0 : (j == 1) ? acc1 : (j == 2) ? acc2 : acc3;
            acc = __builtin_amdgcn_wmma_f32_16x16x4_f32(
                false, a, false, b, (short)0, acc, false, false);
        }
    }

    v8f acc = acc0 + acc1 + acc2 + acc3;
    const float bv = bias[n0 + row];        // bias depends only on N = lane&15
    // C/D layout: VGPR r -> M = r + 8*hi, N = lane&15
    #pragma unroll
    for (int r = 0; r < 8; ++r) {
        const int m = r + 8 * hi;
        dec[(size_t)m * An + n0 + row] = acc[r] + bv;
    }
}

// ---------------------------------------------------------------------------
// Kernel 2: e[b,l] = sum_a v[a]*tanh(feat[b,l,a] + dec[b,a])
// grid (L/8, B), 256 threads (8 waves); one wave per encoder position l.
// dec row + v staged in LDS; float4-coalesced feature reads.
// ---------------------------------------------------------------------------
__global__ __launch_bounds__(256)
void scores_kernel(const float* __restrict__ feat, const float* __restrict__ dec,
                   const float* __restrict__ v, float* __restrict__ e) {
    __shared__ float sdec[An];
    __shared__ float sv[An];
    const int b = blockIdx.y;
    const int tid = threadIdx.x;

    for (int i = tid; i < An; i += 256) {
        sdec[i] = dec[(size_t)b * An + i];
        sv[i]   = v[i];
    }
    __syncthreads();

    const int wave = tid >> 5;
    const int lane = tid & 31;
    const int l = blockIdx.x * 8 + wave;
    const size_t base = ((size_t)b * Ln + l) * An;

    float sum = 0.0f;
    #pragma unroll
    for (int j = 0; j < 4; ++j) {
        const int idx = j * 128 + lane * 4;   // 16B aligned, 512B coalesced/wave
        const float4 f = *reinterpret_cast<const float4*>(feat + base + idx);
        const float4 d = *reinterpret_cast<const float4*>(&sdec[idx]);
        const float4 w = *reinterpret_cast<const float4*>(&sv[idx]);
        sum += w.x * tanhf(f.x + d.x);
        sum += w.y * tanhf(f.y + d.y);
        sum += w.z * tanhf(f.z + d.z);
        sum += w.w * tanhf(f.w + d.w);
    }
    // wave32 butterfly reduction
    #pragma unroll
    for (int m = 16; m >= 1; m >>= 1) sum += __shfl_xor(sum, m, 32);
    if (lane == 0) e[(size_t)b * Ln + l] = sum;
}

// ---------------------------------------------------------------------------
// Kernel 3: masked softmax over L per batch. One block (1024 thr) per b.
// attn = exp(e - max) * mask / sum(exp(e - max) * mask)
// ---------------------------------------------------------------------------
__global__ __launch_bounds__(1024)
void softmax_kernel(const float* __restrict__ e, const float* __restrict__ mask,
                    float* __restrict__ attn) {
    __shared__ float red[1024];
    const int b = blockIdx.x;
    const int tid = threadIdx.x;

    float vals[4];
    float m = -INFINITY;
    #pragma unroll
    for (int i = 0; i < 4; ++i) {
        vals[i] = e[(size_t)b * Ln + tid + i * 1024];
        m = fmaxf(m, vals[i]);
    }
    red[tid] = m;
    __syncthreads();
    for (int s = 512; s > 0; s >>= 1) {
        if (tid < s) red[tid] = fmaxf(red[tid], red[tid + s]);
        __syncthreads();
    }
    m = red[0];
    __syncthreads();

    float ssum = 0.0f;
    #pragma unroll
    for (int i = 0; i < 4; ++i) {
        const int l = tid + i * 1024;
        const float w = __expf(vals[i] - m) * mask[(size_t)b * Ln + l];
        vals[i] = w;
        ssum += w;
    }
    red[tid] = ssum;
    __syncthreads();
    for (int s = 512; s > 0; s >>= 1) {
        if (tid < s) red[tid] += red[tid + s];
        __syncthreads();
    }
    const float inv = 1.0f / red[0];
    #pragma unroll
    for (int i = 0; i < 4; ++i)
        attn[(size_t)b * Ln + tid + i * 1024] = vals[i] * inv;
}

// ---------------------------------------------------------------------------
// Kernel 4: partial context over an L-chunk.
// grid (NCHUNK, B, A/256), 256 threads. partial[chunk][b][a] in workspace.
// ---------------------------------------------------------------------------
__global__ __launch_bounds__(256)
void context_partial_kernel(const float* __restrict__ attn,
                            const float* __restrict__ state,
                            float* __restrict__ partial) {
    __shared__ float sattn[LCHUNK];
    const int chunk = blockIdx.x;
    const int b = blockIdx.y;
    const int tid = threadIdx.x;

    if (tid < LCHUNK) sattn[tid] = attn[(size_t)b * Ln + chunk * LCHUNK + tid];
    __syncthreads();

    const int a = blockIdx.z * 256 + tid;
    const float* __restrict__ sp =
        state + ((size_t)b * Ln + (size_t)chunk * LCHUNK) * An + a;

    float acc = 0.0f;
    #pragma unroll 8
    for (int i = 0; i < LCHUNK; ++i) {
        __builtin_prefetch(sp + (i + 8) * An, 0, 1);   // global_prefetch_b8
        acc += sattn[i] * sp[(size_t)i * An];
    }
    partial[((size_t)chunk * Bn + b) * An + a] = acc;
}

// ---------------------------------------------------------------------------
// Kernel 5: reduce 32 partials -> context. grid (A/256, B), 256 threads.
// ---------------------------------------------------------------------------
__global__ __launch_bounds__(256)
void context_reduce_kernel(const float* __restrict__ partial,
                           float* __restrict__ context) {
    const int b = blockIdx.y;
    const int a = blockIdx.x * 256 + threadIdx.x;
    float acc = 0.0f;
    #pragma unroll
    for (int ch = 0; ch < NCHUNK; ++ch)
        acc += partial[((size_t)ch * Bn + b) * An + a];
    context[(size_t)b * An + a] = acc;
}

// ---------------------------------------------------------------------------
extern "C" void kernel_launch(void* const* d_in, const int* in_sizes, int n_in,
                              void* d_out, int out_size, void* d_ws, size_t ws_size,
                              hipStream_t stream) {
    const float* feat  = (const float*)d_in[0];  // [B,L,1,A]
    const float* h     = (const float*)d_in[1];  // [B,H]
    const float* c     = (const float*)d_in[2];  // [B,H]
    const float* state = (const float*)d_in[3];  // [B,L,1,A]
    const float* mask  = (const float*)d_in[4];  // [B,L]
    const float* v     = (const float*)d_in[5];  // [A]
    const float* W     = (const float*)d_in[6];  // [A,2H]
    const float* bias  = (const float*)d_in[7];  // [A]

    float* context = (float*)d_out;              // [B,A]   = 8192
    float* attn    = (float*)d_out + Bn * An;    // [B,L]   = 65536

    float* ws      = (float*)d_ws;
    float* dec     = ws;                         // 16*512          = 32 KB
    float* e       = dec + Bn * An;              // 16*4096         = 256 KB
    float* partial = e + Bn * Ln;                // 32*16*512       = 1 MB

    // 1) dec = cat(h,c) @ W^T + b   (WMMA f32 16x16x4)
    dec_wmma_kernel<<<dim3(An / 16), dim3(32), 0, stream>>>(h, c, W, bias, dec);

    // 2) additive-attention scores e[b,l]
    scores_kernel<<<dim3(Ln / 8, Bn), dim3(256), 0, stream>>>(feat, dec, v, e);

    // 3) masked softmax -> attn (output)
    softmax_kernel<<<dim3(Bn), dim3(1024), 0, stream>>>(e, mask, attn);

    // 4) context partial sums over L-chunks
    context_partial_kernel<<<dim3(NCHUNK, Bn, An / 256), dim3(256), 0, stream>>>(
        attn, state, partial);

    // 5) reduce partials -> context (output)
    context_reduce_kernel<<<dim3(An / 256, Bn), dim3(256), 0, stream>>>(
        partial, context);
}